// CustomFullyConnectedLayer_75737453297978
// MI455X (gfx1250) — compile-verified
//
#include <hip/hip_runtime.h>

// ---------------------------------------------------------------------------
// Problem constants (from the reference)
// ---------------------------------------------------------------------------
#define NFEAT   4096        // IN_FEATURES == OUT_FEATURES == P == D
#define MROWS   512         // batch rows of x
#define TOPK    3687        // K = ceil(0.9 * 4096 * 4096 / 4096)
#define NITER   50          // Dykstra iterations
#define INV_LR  100.0f      // 1 / ALPHA_LR

// GEMM tiling
#define BM 128
#define BN 128
#define BK 32
#define NTILES (NFEAT / BK)   // 128 k-slabs
#define XS_STRIDE 36          // 32 + 4 pad: float4 alignment + conflict-free frag reads
#define VS_STRIDE 36
#define VS_ROWS   160         // BN + BK - 1 = 159 rounded up

typedef __attribute__((ext_vector_type(2))) float v2f;
typedef __attribute__((ext_vector_type(8))) float v8f;
typedef __attribute__((ext_vector_type(4))) int   v4i;

// CDNA5 async global->LDS copy path (guarded: falls back to register staging)
#if defined(__gfx1250__) && \
    __has_builtin(__builtin_amdgcn_global_load_async_to_lds_b128) && \
    __has_builtin(__builtin_amdgcn_s_wait_asynccnt)
#define USE_ASYNC_LDS 1
typedef __attribute__((address_space(1))) v4i gv4i;   // global int32x4
typedef __attribute__((address_space(3))) v4i lv4i;   // LDS int32x4
#else
#define USE_ASYNC_LDS 0
#endif

// ---------------------------------------------------------------------------
// Kernel 1: Dykstra soft top-k mask.
// The reference's `p` is a uniform scalar that cancels out of y, so:
//   y = x + (K - sum(x))/n ;  u = y + q ;  x = clip(u,0,1) ;  q = u - x
// One workgroup, 256 threads, 16 elements/thread in registers.
// ---------------------------------------------------------------------------
__global__ __launch_bounds__(256) void dykstra_mask_kernel(
    const float* __restrict__ alpha, float* __restrict__ a_out)
{
    __shared__ float red[256];
    const int tid = threadIdx.x;

    float x[16], q[16];
#pragma unroll
    for (int j = 0; j < 16; ++j) {
        x[j] = alpha[tid * 16 + j] * INV_LR;
        q[j] = 0.0f;
    }

    for (int it = 0; it < NITER; ++it) {
        float s = 0.0f;
#pragma unroll
        for (int j = 0; j < 16; ++j) s += x[j];
        red[tid] = s;
        __syncthreads();
        for (int off = 128; off > 0; off >>= 1) {
            if (tid < off) red[tid] += red[tid + off];
            __syncthreads();
        }
        const float sum = red[0];
        __syncthreads();

        const float delta = ((float)TOPK - sum) / (float)NFEAT;
#pragma unroll
        for (int j = 0; j < 16; ++j) {
            float u  = x[j] + delta + q[j];
            float xn = fminf(fmaxf(u, 0.0f), 1.0f);
            q[j] = u - xn;
            x[j] = xn;
        }
    }

#pragma unroll
    for (int j = 0; j < 16; ++j) a_out[tid * 16 + j] = x[j];
}

// ---------------------------------------------------------------------------
// Kernel 2: fused pseudo-diagonal GEMM, software-pipelined.
//   out[m,n] = sum_k x[m,k] * a[(n-k)&4095] * V[(n-k)&4095, k]
// - xs: double-buffered x tile, filled by GLOBAL_LOAD_ASYNC_TO_LDS_B128
//   (pure copy) while WMMAs consume the other buffer.
// - vs: single-buffered a*V band; next slab is prefetched into registers
//   during compute, scaled by a[i], and committed after the read barrier.
// Compute: V_WMMA_F32_16X16X4_F32, 8 waves (2x4), 64x32 macro-tile per wave.
// ---------------------------------------------------------------------------
__global__ __launch_bounds__(256) void pseudodiag_gemm_kernel(
    const float* __restrict__ x, const float* __restrict__ V,
    const float* __restrict__ a, float* __restrict__ out)
{
    __shared__ float xs[2][BM * XS_STRIDE];     // 2 x 18 KB
    __shared__ float vs[VS_ROWS * VS_STRIDE];   // 23 KB

    const int tid   = threadIdx.x;
    const int lane  = tid & 31;
    const int wave  = tid >> 5;
    const int wm    = wave >> 2;                // 0..1
    const int wn    = wave & 3;                 // 0..3
    const int l16   = lane & 15;
    const int lhalf = lane >> 4;

    const int n0 = blockIdx.x * BN;
    const int m0 = blockIdx.y * BM;

    const v8f vzero = {0.f, 0.f, 0.f, 0.f, 0.f, 0.f, 0.f, 0.f};
    v8f acc[4][2];
#pragma unroll
    for (int mt = 0; mt < 4; ++mt)
#pragma unroll
        for (int nt = 0; nt < 2; ++nt) acc[mt][nt] = vzero;

    // Per-thread slot geometry (fixed across slabs)
    const int xrow[4] = { (0*256 + tid) >> 3, (1*256 + tid) >> 3,
                          (2*256 + tid) >> 3, (3*256 + tid) >> 3 };
    const int xkq     = (tid & 7) << 2;

    // ---------------- prologue: fill slab 0 synchronously ----------------
    {
#pragma unroll
        for (int it = 0; it < 4; ++it) {
            float4 t = *(const float4*)(x + (size_t)(m0 + xrow[it]) * NFEAT + xkq);
            *(float4*)(&xs[0][xrow[it] * XS_STRIDE + xkq]) = t;
        }
        const int iBase = (n0 - (BK - 1) + 2 * NFEAT) & (NFEAT - 1);
#pragma unroll
        for (int it = 0; it < 5; ++it) {
            int slot = it * 256 + tid;
            int row  = slot >> 3;
            int kq   = (slot & 7) << 2;
            int ig   = (iBase + row) & (NFEAT - 1);
            float av = a[ig];
            float4 t = *(const float4*)(V + (size_t)ig * NFEAT + kq);
            float4 w;
            w.x = av * t.x; w.y = av * t.y; w.z = av * t.z; w.w = av * t.w;
            *(float4*)(&vs[row * VS_STRIDE + kq]) = w;
        }
    }
    __syncthreads();

    // ---------------- main pipelined loop over 128 k-slabs ----------------
    float4 vpre[5];
    float  apre[5];
#if !USE_ASYNC_LDS
    float4 xpre[4];
#endif

    for (int t = 0; t < NTILES; ++t) {
        const int  k0      = t * BK;
        const int  cur     = t & 1;
        const int  nxt     = cur ^ 1;
        const bool hasNext = (t + 1 < NTILES);

        // -- issue next-slab fetches before compute (latency hiding) --
        if (hasNext) {
            const int kn = k0 + BK;
#if USE_ASYNC_LDS
#pragma unroll
            for (int it = 0; it < 4; ++it) {
                const float* gp = x + (size_t)(m0 + xrow[it]) * NFEAT + kn + xkq;
                float* lp = &xs[nxt][xrow[it] * XS_STRIDE + xkq];
                __builtin_amdgcn_global_load_async_to_lds_b128(
                    (gv4i*)gp, (lv4i*)lp, 0, 0);
            }
#else
#pragma unroll
            for (int it = 0; it < 4; ++it)
                xpre[it] = *(const float4*)(x + (size_t)(m0 + xrow[it]) * NFEAT + kn + xkq);
#endif
            const int iBaseN = (n0 - kn - (BK - 1) + 2 * NFEAT) & (NFEAT - 1);
#pragma unroll
            for (int it = 0; it < 5; ++it) {
                int slot = it * 256 + tid;
                int row  = slot >> 3;
                int kq   = (slot & 7) << 2;
                int ig   = (iBaseN + row) & (NFEAT - 1);
                apre[it] = a[ig];
                vpre[it] = *(const float4*)(V + (size_t)ig * NFEAT + kn + kq);
            }
        }

        // -- compute current slab: 8 K-steps of V_WMMA_F32_16X16X4_F32 --
#pragma unroll
        for (int ks = 0; ks < 8; ++ks) {
            const int kl = ks * 4 + lhalf * 2;

            v2f af[4];
#pragma unroll
            for (int mt = 0; mt < 4; ++mt) {
                const float* p = &xs[cur][(wm * 64 + mt * 16 + l16) * XS_STRIDE + kl];
                v2f tt = {p[0], p[1]};
                af[mt] = tt;
            }
            v2f bf[2];
#pragma unroll
            for (int nt = 0; nt < 2; ++nt) {
                const int nl = wn * 32 + nt * 16 + l16;
                float b0 = vs[(nl - kl     + (BK - 1)) * VS_STRIDE + kl];
                float b1 = vs[(nl - kl - 1 + (BK - 1)) * VS_STRIDE + kl + 1];
                v2f tt = {b0, b1};
                bf[nt] = tt;
            }
#pragma unroll
            for (int mt = 0; mt < 4; ++mt)
#pragma unroll
                for (int nt = 0; nt < 2; ++nt)
                    acc[mt][nt] = __builtin_amdgcn_wmma_f32_16x16x4_f32(
                        false, af[mt], false, bf[nt],
                        (short)0, acc[mt][nt], false, false);
        }

        __syncthreads();   // all waves done reading vs (and xs[cur])

        // -- commit next slab: scale+store vs band; ensure async xs landed --
        if (hasNext) {
#pragma unroll
            for (int it = 0; it < 5; ++it) {
                int slot = it * 256 + tid;
                int row  = slot >> 3;
                int kq   = (slot & 7) << 2;
                float4 w;
                w.x = apre[it] * vpre[it].x; w.y = apre[it] * vpre[it].y;
                w.z = apre[it] * vpre[it].z; w.w = apre[it] * vpre[it].w;
                *(float4*)(&vs[row * VS_STRIDE + kq]) = w;
            }
#if USE_ASYNC_LDS
            __builtin_amdgcn_s_wait_asynccnt(0);
#else
#pragma unroll
            for (int it = 0; it < 4; ++it)
                *(float4*)(&xs[nxt][xrow[it] * XS_STRIDE + xkq]) = xpre[it];
#endif
        }
        __syncthreads();
    }

    // ---------------- epilogue: C/D layout -> global ----------------
#pragma unroll
    for (int mt = 0; mt < 4; ++mt) {
#pragma unroll
        for (int nt = 0; nt < 2; ++nt) {
            const int Ng = n0 + wn * 32 + nt * 16 + l16;
#pragma unroll
            for (int v = 0; v < 8; ++v) {
                const int Mg = m0 + wm * 64 + mt * 16 + v + 8 * lhalf;
                out[(size_t)Mg * NFEAT + Ng] = acc[mt][nt][v];
            }
        }
    }
}

// ---------------------------------------------------------------------------
// Host-side launcher
// d_in[0] = x (512*4096 f32), d_in[1] = V (4096*4096 f32), d_in[2] = alpha (4096 f32)
// d_out   = out (512*4096 f32); d_ws: first 16 KB holds the mask vector a[4096]
// ---------------------------------------------------------------------------
extern "C" void kernel_launch(void* const* d_in, const int* in_sizes, int n_in,
                              void* d_out, int out_size, void* d_ws, size_t ws_size,
                              hipStream_t stream) {
    const float* xin   = (const float*)d_in[0];
    const float* V     = (const float*)d_in[1];
    const float* alpha = (const float*)d_in[2];
    float* out = (float*)d_out;
    float* a   = (float*)d_ws;

    dykstra_mask_kernel<<<1, 256, 0, stream>>>(alpha, a);

    dim3 grid(NFEAT / BN, MROWS / BM);   // 32 x 4
    pseudodiag_gemm_kernel<<<grid, 256, 0, stream>>>(xin, V, a, out);

    (void)in_sizes; (void)n_in; (void)out_size; (void)ws_size;
}